// VFPhi_5549097747173
// MI455X (gfx1250) — compile-verified
//
#include <hip/hip_runtime.h>

// Native vector type so __builtin_nontemporal_load lowers to a single
// global_load_b128 with the NT temporal hint.
typedef __attribute__((ext_vector_type(4))) float v4f;

__global__ __launch_bounds__(256) void zone_gather_kernel(
    const v4f*  __restrict__ z1,    // [B][4] f32
    const v4f*  __restrict__ z2,    // [B][4] f32
    const float* __restrict__ dir,  // [B][8] f32
    float* __restrict__ out,        // [B]    f32
    int B)
{
    int i = blockIdx.x * blockDim.x + threadIdx.x;
    if (i >= B) return;

    // 16B coalesced, non-temporal (stream-once, > L2 working set)
    v4f a = __builtin_nontemporal_load(&z1[i]);
    v4f b = __builtin_nontemporal_load(&z2[i]);

    float vx = b.x - a.x;
    float vy = -(b.z - a.z);

    // degrees(atan2(vy, vx)) ; 180/pi
    float phi = atan2f(vy, vx) * 57.29577951308232f;

    // truncation toward zero, matching .astype(int32)
    int phi_long = (int)phi;

    // jnp.mod(90 - phi_long, 360): non-negative result
    int pcs = (90 - phi_long) % 360;
    pcs += (pcs >> 31) & 360;   // add 360 iff negative (branchless)

    // floor((pcs + 22.5)/45) for pcs in [0,359] -> value in [0,8]; %8 == &7
    int zone = (int)(((float)pcs + 22.5f) * (1.0f / 45.0f)) & 7;

    // 4B gather within this row's 32B; consecutive threads cover consecutive
    // rows, so the cacheline footprint is identical to a full-row load but
    // with 1 VMEM op and 1 VGPR instead of 8.
    float v = __builtin_nontemporal_load(&dir[(size_t)i * 8 + zone]);

    __builtin_nontemporal_store(v, &out[i]);
}

extern "C" void kernel_launch(void* const* d_in, const int* in_sizes, int n_in,
                              void* d_out, int out_size, void* d_ws, size_t ws_size,
                              hipStream_t stream) {
    (void)n_in; (void)d_ws; (void)ws_size;

    const v4f*   z1  = (const v4f*)d_in[0];   // z_1: [B,4] f32
    const v4f*   z2  = (const v4f*)d_in[1];   // z_2: [B,4] f32
    const float* dir = (const float*)d_in[2]; // dir: [B,8] f32
    float*       out = (float*)d_out;         // [B] f32

    const int B = in_sizes[0] / 4;            // 4,000,000
    const int threads = 256;                  // 8 wave32s per block
    const int blocks  = (B + threads - 1) / threads;

    zone_gather_kernel<<<blocks, threads, 0, stream>>>(z1, z2, dir, out, B);
}